// FBatchnorm2d_57380763074661
// MI455X (gfx1250) — compile-verified
//
#include <hip/hip_runtime.h>
#include <hip/hip_bf16.h>

// FBatchnorm2d with Fourier stats, MI455X (gfx1250, wave32).
//
// Math: for real x, with sx = ifft2(x) (1/(HW) normalization):
//   sum_{h,w} sx          = x[0,0]                      -> mean_c = S1_c/(bs*H*W)
//   sum_{h,w} Re(sx)^2    = (S2 + S3) / (2*H*W)
//     where S2 = sum x^2,  S3 = sum x[u,v]*x[(-u)%H,(-v)%W]  (real)
//   E[sx^2]_c = (S2_c + S3_c) / (2*bs*H^2*W^2),  var_c = E[sx^2]_c - mean_c^2
// Then: rm = 0.8*rmean + 0.2*mean ; rv = 0.8*rvar + 0.2*var
//       out = gamma/sqrt(rv+eps) * (x - rm) + beta   (elementwise on ORIGINAL x)
//
// Pipeline (bandwidth-bound; ~288 MB HBM traffic @ 23.3 TB/s ~ 12 us):
//   K1 stats:    single full read of x via b128 loads. Row pairs (h, H-h) per
//                block; both rows staged in LDS (ds_store_b128), mirror read
//                reversed (stride -1 over 64 banks: conflict-free). Block
//                (s2,s3) reduced with V_WMMA_F32_16X16X4_F32.
//   K2 finalize: tiny; fp64 sums of 512 partials/channel -> scale/shift.
//   K3 norm:     b128 read x (L2-resident), FMA, non-temporal b128 store.

#define BS   32
#define CH   3
#define HH   512
#define WW   512
#define NPAIRS (BS * 257)   // per channel: hp in [0,256], rows (hp, 512-hp)
#define NBLK  512           // stats blocks per channel

typedef __attribute__((ext_vector_type(2))) float v2f;
typedef __attribute__((ext_vector_type(4))) float v4f;
typedef __attribute__((ext_vector_type(8))) float v8f;

__global__ __launch_bounds__(256)
void fbn_stats_kernel(const float* __restrict__ x, float* __restrict__ partials)
{
    const int c   = blockIdx.y;
    const int tid = threadIdx.x;

    __shared__ float lds[2 * WW];  // [0,512): row A, [512,1024): row B (4 KB)
    __shared__ float red[32];      // 8 waves * 4 slots

    const int q = tid & 127;       // quad index within a row
    const int r = tid >> 7;        // 0 -> stage rowA, 1 -> stage rowB

    float s2 = 0.0f, s3 = 0.0f;

    for (int p = blockIdx.x; p < NPAIRS; p += gridDim.x) {
        const int b  = p / 257;
        const int hp = p - b * 257;
        const int h2 = (HH - hp) & (HH - 1);          // 0->0, 256->256
        const size_t chanBase = ((size_t)b * CH + c) * (size_t)HH * WW;
        const float* rowA = x + chanBase + (size_t)hp * WW;
        const float* rowB = x + chanBase + (size_t)h2 * WW;

        // prefetch next pair (global_prefetch_b8); 256 lanes * 8B span = 2KB row
        const int pn = p + gridDim.x;
        if (pn < NPAIRS) {
            const int bn  = pn / 257;
            const int hpn = pn - bn * 257;
            const int h2n = (HH - hpn) & (HH - 1);
            const size_t nb = ((size_t)bn * CH + c) * (size_t)HH * WW;
            __builtin_prefetch(x + nb + (size_t)hpn * WW + tid * 2, 0, 0);
            __builtin_prefetch(x + nb + (size_t)h2n * WW + tid * 2, 0, 0);
        }

        // stage both rows into LDS: global_load_b128 + ds_store_b128
        const v4f quad = reinterpret_cast<const v4f*>(r ? rowB : rowA)[q];
        reinterpret_cast<v4f*>(lds)[(r << 7) + q] = quad;
        __syncthreads();

        const bool self = (h2 == hp);   // uniform across block
        #pragma unroll
        for (int k = 0; k < 2; ++k) {
            const int   w  = tid + (k << 8);
            const float a  = lds[w];
            const int   wm = (WW - w) & (WW - 1);
            const float bm = lds[WW + wm];       // reversed read, no conflicts
            if (self) {
                s2 += a * a;
                s3 += a * bm;
            } else {
                const float bf = lds[WW + w];
                s2 += a * a + bf * bf;
                s3 += 2.0f * a * bm;             // both cross terms are equal
            }
        }
        __syncthreads();
    }

    // ---- wave-level (s2,s3) reduction via V_WMMA_F32_16X16X4_F32 ----
    // A (16x4): lane m in [0,15]: A[m][0]=s2, A[m][1]=s3 (VGPR0/1);
    //           lane m+16:        A[m][2]=s2, A[m][3]=s3.
    // B (4x16): B[0][0]=B[2][0]=1, B[1][1]=B[3][1]=1, else 0
    //   -> D[m][0] = s2(m)+s2(m+16), D[m][1] = s3(m)+s3(m+16).
    // Summing the 8 D VGPRs in-lane leaves wave totals split across lanes
    // {0,16} (s2) and {1,17} (s3). EXEC is all ones here (no divergence).
    const int lane = tid & 31;
    const int wave = tid >> 5;
    v2f A; A.x = s2; A.y = s3;
    v2f B;
    B.x = (lane == 0 || lane == 16) ? 1.0f : 0.0f;
    B.y = (lane == 1 || lane == 17) ? 1.0f : 0.0f;
    v8f Cz = {};
    v8f D = __builtin_amdgcn_wmma_f32_16x16x4_f32(
        /*neg_a=*/false, A, /*neg_b=*/false, B,
        /*c_mod=*/(short)0, Cz, /*reuse_a=*/false, /*reuse_b=*/false);

    float v = 0.0f;
    #pragma unroll
    for (int i = 0; i < 8; ++i) v += D[i];

    // lanes {0,1,16,17} <=> (lane & 14) == 0 ; slot: 0->0, 16->1, 1->2, 17->3
    if ((lane & 14) == 0) {
        const int slot = ((lane & 1) << 1) | (lane >> 4);
        red[(wave << 2) + slot] = v;
    }
    __syncthreads();

    if (tid == 0) {
        float bs2 = 0.0f, bs3 = 0.0f;
        #pragma unroll
        for (int w2 = 0; w2 < 8; ++w2) {
            bs2 += red[w2 * 4 + 0] + red[w2 * 4 + 1];
            bs3 += red[w2 * 4 + 2] + red[w2 * 4 + 3];
        }
        const size_t slot = ((size_t)c * gridDim.x + blockIdx.x) * 2;
        partials[slot + 0] = bs2;
        partials[slot + 1] = bs3;
    }
}

__global__ void fbn_finalize_kernel(const float* __restrict__ x,
                                    const float* __restrict__ gamma,
                                    const float* __restrict__ beta,
                                    const float* __restrict__ rmean,
                                    const float* __restrict__ rvar,
                                    const float* __restrict__ partials,
                                    float* __restrict__ scaleshift)
{
    const int c = threadIdx.x;
    if (c >= CH) return;

    double S2 = 0.0, S3 = 0.0;
    for (int b = 0; b < NBLK; ++b) {
        S2 += (double)partials[((size_t)c * NBLK + b) * 2 + 0];
        S3 += (double)partials[((size_t)c * NBLK + b) * 2 + 1];
    }
    double S1 = 0.0;
    for (int b = 0; b < BS; ++b)
        S1 += (double)x[((size_t)b * CH + c) * (size_t)HH * WW];  // x[b,c,0,0]

    const double cnt  = (double)BS * HH * WW;                       // bs*H*W
    const double mean = S1 / cnt;
    const double e2   = (S2 + S3) /
                        (2.0 * (double)BS * (double)HH * HH * (double)WW * WW);
    const double var  = e2 - mean * mean;

    const float rm = rmean[c] * 0.8f + 0.2f * (float)mean;
    const float rv = rvar[c]  * 0.8f + 0.2f * (float)var;
    const float inv = rsqrtf(rv + 1e-5f);
    const float scale = gamma[c] * inv;
    scaleshift[c]      = scale;
    scaleshift[CH + c] = beta[c] - scale * rm;
}

__global__ __launch_bounds__(256)
void fbn_norm_kernel(const float* __restrict__ x,
                     const float* __restrict__ ss,
                     float* __restrict__ out, long n4)
{
    const long i = (long)blockIdx.x * 256 + threadIdx.x;
    if (i >= n4) return;
    const long e = i * 4;
    const int  c = (int)((e >> 18) % CH);        // H*W = 2^18; quad is intra-channel
    const float scale = ss[c];
    const float shift = ss[CH + c];
    const v4f xv = reinterpret_cast<const v4f*>(x)[i];
    v4f o;
    o.x = fmaf(scale, xv.x, shift);
    o.y = fmaf(scale, xv.y, shift);
    o.z = fmaf(scale, xv.z, shift);
    o.w = fmaf(scale, xv.w, shift);
    // out is never re-read: non-temporal store keeps x resident in L2
    __builtin_nontemporal_store(o, reinterpret_cast<v4f*>(out) + i);
}

extern "C" void kernel_launch(void* const* d_in, const int* in_sizes, int n_in,
                              void* d_out, int out_size, void* d_ws, size_t ws_size,
                              hipStream_t stream)
{
    const float* x     = (const float*)d_in[0];
    const float* gamma = (const float*)d_in[1];
    const float* beta  = (const float*)d_in[2];
    const float* rmean = (const float*)d_in[3];
    const float* rvar  = (const float*)d_in[4];
    float*       out   = (float*)d_out;

    float* ws       = (float*)d_ws;
    float* partials = ws;                    // 2 * CH * NBLK = 3072 floats
    float* ss       = ws + 2 * CH * NBLK;    // scale[3], shift[3]

    fbn_stats_kernel<<<dim3(NBLK, CH), 256, 0, stream>>>(x, partials);
    fbn_finalize_kernel<<<1, 32, 0, stream>>>(x, gamma, beta, rmean, rvar,
                                              partials, ss);
    const long n4 = (long)BS * CH * HH * WW / 4;          // 6,291,456
    const int  nblocks = (int)((n4 + 255) / 256);         // 24,576
    fbn_norm_kernel<<<nblocks, 256, 0, stream>>>(x, ss, out, n4);
}